// DualTPP_20693152432316
// MI455X (gfx1250) — compile-verified
//
#include <hip/hip_runtime.h>

// DualTPP reconciliation: H is block-diagonal (8 blocks of 12x12 per sample),
// each block = diag(w1) + rank-3 update -> closed-form Woodbury solve.
// T=96, K6=6, K12=12 -> blocks of 12 rows aligned with one 12-agg bin and two 6-agg bins.

__global__ __launch_bounds__(256) void dualtpp_woodbury_kernel(
    const float* __restrict__ mu1,  const float* __restrict__ sd1,
    const float* __restrict__ mu6,  const float* __restrict__ sd6,
    const float* __restrict__ mu12, const float* __restrict__ sd12,
    const float* __restrict__ norm1, const float* __restrict__ norm6,
    const float* __restrict__ norm12,
    float* __restrict__ out, int nblk)
{
    const int g = blockIdx.x * 256 + threadIdx.x;   // one thread = one 12x12 block
    if (g >= nblk) return;
    const int b = g >> 3;       // sample
    const int j = g & 7;        // block within sample (0..7)

    const size_t off1  = (size_t)b * 96 + (size_t)j * 12;  // 12 fine elements
    const size_t off6  = (size_t)b * 16 + (size_t)j * 2;   // 2 mid elements
    const size_t off12 = (size_t)b * 8  + (size_t)j;       // 1 coarse element

    // CDNA5 speculative prefetch of the streaming inputs one WG-slab ahead
    // (lowered to global_prefetch_b8; dropped silently if past the buffer).
    __builtin_prefetch(mu1 + off1 + 96 * 32, 0, 0);
    __builtin_prefetch(sd1 + off1 + 96 * 32, 0, 0);

    // ---- loads (all naturally aligned: 48B / 8B / 4B offsets) ----
    const float4* m1p = (const float4*)(mu1 + off1);
    const float4* s1p = (const float4*)(sd1 + off1);
    const float4 mA = m1p[0], mB = m1p[1], mC = m1p[2];
    const float4 sA = s1p[0], sB = s1p[1], sC = s1p[2];

    const float2 m6v = *(const float2*)(mu6 + off6);
    const float2 s6v = *(const float2*)(sd6 + off6);
    const float  m12v = mu12[off12];
    const float  s12v = sd12[off12];

    const float c6  = norm1[0] / norm6[0];
    const float c12 = norm1[0] / norm12[0];

    // ---- per-block scalars ----
    const float s6a2 = s6v.x * s6v.x;
    const float s6b2 = s6v.y * s6v.y;
    const float s122 = s12v * s12v;
    const float w6a = 1.0f / s6a2, w6b = 1.0f / s6b2, w12 = 1.0f / s122;

    const float add12 = c12 * w12 * m12v;
    const float addLo = c6 * w6a * m6v.x + add12;  // s2_i * addLo folds the A6^T/A12^T rhs terms
    const float addHi = c6 * w6b * m6v.y + add12;

    const float mus[12] = {mA.x, mA.y, mA.z, mA.w, mB.x, mB.y, mB.z, mB.w, mC.x, mC.y, mC.z, mC.w};
    const float sds[12] = {sA.x, sA.y, sA.z, sA.w, sB.x, sB.y, sB.z, sB.w, sC.x, sC.y, sC.z, sC.w};

    float s2[12], y[12];
    float p = 0.0f, q = 0.0f, t1 = 0.0f, t2 = 0.0f;
    #pragma unroll
    for (int i = 0; i < 6; ++i) {
        s2[i] = sds[i] * sds[i];            // D^-1 diagonal (stds^2)
        y[i]  = mus[i] + s2[i] * addLo;     // y = D^-1 * rhs
        p  += s2[i];
        t1 += y[i];
    }
    #pragma unroll
    for (int i = 6; i < 12; ++i) {
        s2[i] = sds[i] * sds[i];
        y[i]  = mus[i] + s2[i] * addHi;
        q  += s2[i];
        t2 += y[i];
    }

    // ---- 3x3 capacitance solve: M = C^-1 + U^T D^-1 U, M12 == 0 ----
    const float inv_c6sq  = 1.0f / (c6 * c6);
    const float inv_c12sq = 1.0f / (c12 * c12);
    const float A11 = s6a2 * inv_c6sq + p;
    const float A22 = s6b2 * inv_c6sq + q;
    const float A33 = s122 * inv_c12sq + p + q;

    const float k1 = p / A11;
    const float k2 = q / A22;
    const float z3 = (t1 + t2 - k1 * t1 - k2 * t2) / (A33 - k1 * p - k2 * q);
    const float z1 = (t1 - p * z3) / A11;
    const float z2 = (t2 - q * z3) / A22;

    const float corrLo = z1 + z3;   // (U z)_i for first half
    const float corrHi = z2 + z3;   // (U z)_i for second half

    // ---- x = y - D^-1 U z ----
    float x[12];
    #pragma unroll
    for (int i = 0; i < 6; ++i)  x[i] = y[i] - s2[i] * corrLo;
    #pragma unroll
    for (int i = 6; i < 12; ++i) x[i] = y[i] - s2[i] * corrHi;

    float4* op = (float4*)(out + off1);
    op[0] = make_float4(x[0], x[1], x[2],  x[3]);
    op[1] = make_float4(x[4], x[5], x[6],  x[7]);
    op[2] = make_float4(x[8], x[9], x[10], x[11]);
}

extern "C" void kernel_launch(void* const* d_in, const int* in_sizes, int n_in,
                              void* d_out, int out_size, void* d_ws, size_t ws_size,
                              hipStream_t stream) {
    const float* mu1   = (const float*)d_in[0];
    const float* sd1   = (const float*)d_in[1];
    const float* mu6   = (const float*)d_in[2];
    const float* sd6   = (const float*)d_in[3];
    const float* mu12  = (const float*)d_in[4];
    const float* sd12  = (const float*)d_in[5];
    const float* norm1 = (const float*)d_in[6];
    const float* norm6 = (const float*)d_in[7];
    const float* norm12= (const float*)d_in[8];
    float* out = (float*)d_out;

    const int B    = in_sizes[0] / 96;  // 8192
    const int nblk = B * 8;             // 65536 independent 12x12 solves
    const int threads = 256;
    const int blocks  = (nblk + threads - 1) / threads;

    dualtpp_woodbury_kernel<<<blocks, threads, 0, stream>>>(
        mu1, sd1, mu6, sd6, mu12, sd12, norm1, norm6, norm12, out, nblk);
}